// AttentionGINConv_83743272337691
// MI455X (gfx1250) — compile-verified
//
#include <hip/hip_runtime.h>

typedef __attribute__((ext_vector_type(2))) float v2f;
typedef __attribute__((ext_vector_type(8))) float v8f;

#define C_DIM 64

// ---------------------------------------------------------------------------
// Pass 0: zero the output accumulator, init segment-max to -inf, denom to 0.
// ---------------------------------------------------------------------------
__global__ __launch_bounds__(256) void agc_init_kernel(float* __restrict__ out,
                                                       int* __restrict__ segmax_bits,
                                                       float* __restrict__ denom,
                                                       long long totalOut, int N) {
    long long idx = (long long)blockIdx.x * blockDim.x + threadIdx.x;
    if (idx < totalOut) out[idx] = 0.0f;
    if (idx < N) {
        segmax_bits[idx] = (int)0xFF800000;  // -inf bit pattern
        denom[idx] = 0.0f;
    }
}

// ---------------------------------------------------------------------------
// Pass A: per-node projections a[n] = x[n].w[0:64], b[n] = x[n].w[64:128]
// via V_WMMA_F32_16X16X4_F32. One wave handles 16 nodes; 16 accumulating
// WMMAs cover K=64. B-matrix column 0 = w1, column 1 = w2, rest zero.
// ---------------------------------------------------------------------------
__global__ __launch_bounds__(256) void agc_node_proj_kernel(const float* __restrict__ x,
                                                            const float* __restrict__ att_w,
                                                            float* __restrict__ a_arr,
                                                            float* __restrict__ b_arr,
                                                            int N) {
    const int lane = threadIdx.x & 31;
    const int waveInBlock = threadIdx.x >> 5;
    const int waveGlobal = blockIdx.x * (blockDim.x >> 5) + waveInBlock;
    const int base = waveGlobal * 16;
    if (base >= N) return;                       // wave-uniform exit: EXEC stays full

    const int m = lane & 15;                     // A row this lane supplies
    const int n = lane & 15;                     // D/B column this lane supplies
    const int koff = (lane < 16) ? 0 : 2;        // K sub-slot per ISA A/B layout

    int ridx = base + m;
    if (ridx >= N) ridx = N - 1;                 // clamp (stores are guarded below)
    const float* row = x + (size_t)ridx * C_DIM;

    // B column pointer: col 0 -> w[0:64], col 1 -> w[64:128]; others masked to 0.
    const float msk = (n < 2) ? 1.0f : 0.0f;
    const float* wcol = att_w + ((n < 2) ? n : 0) * C_DIM;

    v8f acc = {};
#pragma unroll
    for (int kk = 0; kk < 16; ++kk) {
        const int k = kk * 4 + koff;
        v2f av;
        av.x = row[k];
        av.y = row[k + 1];
        v2f bv;
        bv.x = wcol[k] * msk;
        bv.y = wcol[k + 1] * msk;
        acc = __builtin_amdgcn_wmma_f32_16x16x4_f32(false, av, false, bv,
                                                    (short)0, acc, false, false);
    }

    // D layout: lanes 0-15 hold rows 0..7 (col = lane); lanes 16-31 rows 8..15.
    if (n < 2) {
        float* dbuf = (n == 0) ? a_arr : b_arr;
        const int mbase = base + ((lane >= 16) ? 8 : 0);
#pragma unroll
        for (int v = 0; v < 8; ++v) {
            const int node = mbase + v;
            if (node < N) dbuf[node] = acc[v];
        }
    }
}

// ---------------------------------------------------------------------------
// Pass B: per-edge logit + atomic float segment-max (signed-max / unsigned-min
// bit trick; segmax initialized to -inf; no NaNs expected).
// ---------------------------------------------------------------------------
__global__ __launch_bounds__(256) void agc_logit_kernel(const int* __restrict__ src,
                                                        const int* __restrict__ dst,
                                                        const float* __restrict__ a_arr,
                                                        const float* __restrict__ b_arr,
                                                        const float* __restrict__ att_b,
                                                        float* __restrict__ logit,
                                                        float* __restrict__ segmax,
                                                        int E) {
    const int e = blockIdx.x * blockDim.x + threadIdx.x;
    if (e >= E) return;
    const int s = src[e];
    const int d = dst[e];
    const float l = a_arr[d] + b_arr[s] + att_b[0];
    logit[e] = l;
    if (l >= 0.0f)
        atomicMax((int*)(segmax + d), __float_as_int(l));
    else
        atomicMin((unsigned int*)(segmax + d), __float_as_uint(l));
}

// ---------------------------------------------------------------------------
// Pass C: ex = exp(logit - segmax[dst]) (in-place), atomic-add into denom.
// ---------------------------------------------------------------------------
__global__ __launch_bounds__(256) void agc_exp_kernel(const int* __restrict__ dst,
                                                      float* __restrict__ logit_ex,
                                                      const float* __restrict__ segmax,
                                                      float* __restrict__ denom,
                                                      int E) {
    const int e = blockIdx.x * blockDim.x + threadIdx.x;
    if (e >= E) return;
    const int d = dst[e];
    const float ex = __expf(logit_ex[e] - segmax[d]);
    logit_ex[e] = ex;
    atomicAdd(denom + d, ex);
}

// ---------------------------------------------------------------------------
// Pass D: out[dst] += alpha * x[src]. One wave covers one edge's 64 channels
// (2 channels / lane, float2 gather from L2-resident x, f32 atomics into out).
// ---------------------------------------------------------------------------
__global__ __launch_bounds__(256) void agc_aggregate_kernel(const int* __restrict__ src,
                                                            const int* __restrict__ dst,
                                                            const float* __restrict__ x,
                                                            const float* __restrict__ ex,
                                                            const float* __restrict__ denom,
                                                            float* __restrict__ out,
                                                            int E) {
    const long long t = (long long)blockIdx.x * blockDim.x + threadIdx.x;
    const long long e = t >> 5;            // 32 lanes -> 64 channels per edge
    if (e >= E) return;
    const int c2 = ((int)t & 31) * 2;
    const int s = src[e];
    const int d = dst[e];
    const float alpha = ex[e] / denom[d];
    const float2 xv = *reinterpret_cast<const float2*>(x + (size_t)s * C_DIM + c2);
    float* op = out + (size_t)d * C_DIM + c2;
    atomicAdd(op, xv.x * alpha);
    atomicAdd(op + 1, xv.y * alpha);
}

// ---------------------------------------------------------------------------
extern "C" void kernel_launch(void* const* d_in, const int* in_sizes, int n_in,
                              void* d_out, int out_size, void* d_ws, size_t ws_size,
                              hipStream_t stream) {
    const float* x = (const float*)d_in[0];
    const int* edge_index = (const int*)d_in[1];
    const float* att_w = (const float*)d_in[2];
    const float* att_b = (const float*)d_in[3];

    const int N = in_sizes[0] / C_DIM;
    const int E = in_sizes[1] / 2;
    const int* src = edge_index;        // edge_index[0, :]
    const int* dst = edge_index + E;    // edge_index[1, :]
    float* out = (float*)d_out;

    // Workspace layout (floats): logit/ex [E] | a [N] | b [N] | segmax [N] | denom [N]
    float* ws = (float*)d_ws;
    float* logit  = ws;
    float* a_arr  = logit + E;
    float* b_arr  = a_arr + N;
    float* segmax = b_arr + N;
    float* denom  = segmax + N;

    const long long totalOut = (long long)N * C_DIM;
    const int ib = 256;
    const int ig = (int)((totalOut + ib - 1) / ib);
    agc_init_kernel<<<ig, ib, 0, stream>>>(out, (int*)segmax, denom, totalOut, N);

    const int nodeWaves = (N + 15) / 16;          // 16 nodes per wave
    const int projBlocks = (nodeWaves + 7) / 8;   // 8 waves per 256-thread block
    agc_node_proj_kernel<<<projBlocks, 256, 0, stream>>>(x, att_w, a_arr, b_arr, N);

    const int eg = (E + 255) / 256;
    agc_logit_kernel<<<eg, 256, 0, stream>>>(src, dst, a_arr, b_arr, att_b, logit, segmax, E);
    agc_exp_kernel<<<eg, 256, 0, stream>>>(dst, logit, segmax, denom, E);

    const long long aggThreads = (long long)E * 32;
    const int ag = (int)((aggThreads + 255) / 256);
    agc_aggregate_kernel<<<ag, 256, 0, stream>>>(src, dst, x, logit, denom, out, E);
}